// flo_map_1752346656963
// MI455X (gfx1250) — compile-verified
//
#include <hip/hip_runtime.h>
#include <math.h>

typedef __attribute__((ext_vector_type(2))) float v2f;
typedef __attribute__((ext_vector_type(8))) float v8f;

#define PI_F    3.14159265358979323846f
#define ANGLE_F 0.78539816339744830962f /* pi/4 */

static constexpr int B_ = 4, C_ = 128, H_ = 128, W_ = 256;
static constexpr int HW = H_ * W_;           // 32768
static constexpr int KTAPS = 9;
static constexpr int TILES_PER_IMG = HW / 16; // 2048

// ---------------------------------------------------------------------------
// Kernel 1: precompute gather index table idx[k][h][w] = iy*W + ix
// (batch/channel independent; pure f32 trig matching the JAX reference)
// ---------------------------------------------------------------------------
__global__ void sphere_idx_kernel(int* __restrict__ g_idx) {
    int tid = blockIdx.x * blockDim.x + threadIdx.x;
    if (tid >= KTAPS * HW) return;
    int k   = tid / HW;
    int rem = tid - k * HW;
    int h   = rem / W_;
    int w   = rem - h * W_;
    int kr = k / 3, kc = k - kr * 3;

    float dy = (float)(kr - 1) * (ANGLE_F * 0.5f);
    float dx = (float)(kc - 1) * (ANGLE_F * 0.5f);
    float ty = tanf(dy), tx = tanf(dx);
    float rho   = sqrtf(tx * tx + ty * ty);
    float cang  = atanf(rho);
    float rho_s = (rho == 0.0f) ? 1.0f : rho;
    float sin_c = sinf(cang), cos_c = cosf(cang);

    float phi = (0.5f - ((float)h + 0.5f) / (float)H_) * PI_F;
    float lam = (((float)w + 0.5f) / (float)W_ - 0.5f) * 2.0f * PI_F;
    float sp = sinf(phi), cp = cosf(phi);

    float t = cos_c * sp + ty * sin_c * cp / rho_s;
    t = fminf(1.0f, fmaxf(-1.0f, t));
    float lat = asinf(t);
    float lon = lam + atan2f(tx * sin_c, rho_s * cp * cos_c - ty * sp * sin_c);

    int iy = (int)rintf((0.5f - lat / PI_F) * (float)H_ - 0.5f);
    iy = min(H_ - 1, max(0, iy));
    int ix = (int)rintf((lon / (2.0f * PI_F) + 0.5f) * (float)W_ - 0.5f);
    ix = ((ix % W_) + W_) % W_;

    g_idx[tid] = iy * W_ + ix;
}

// ---------------------------------------------------------------------------
// Kernel 2: gather + O=1 channel reduction via V_WMMA_F32_16X16X4_F32.
// One wave == one 16-pixel tile (16 consecutive w in one row).
//   A (16x4 f32)  = weights, identical in every M row  -> lane-half uniform
//   B (4x16 f32)  = gathered features, N = pixel (lane%16), K = channel chunk
//   D rows all equal -> acc[0] of lane n is the output for pixel n.
// ---------------------------------------------------------------------------
__global__ __launch_bounds__(256) void sphere_conv_wmma(
    const float* __restrict__ feature,  // [B,C,H,W]
    const float* __restrict__ weight,   // [1,C,3,3]
    const int*   __restrict__ g_idx,    // [9,H,W]
    float*       __restrict__ out)      // [B,1,2H,2W]
{
    // Stage weights channel-major in LDS: wlds[k*C + c] = weight[c*9 + k]
    __shared__ __align__(16) float wlds[KTAPS * C_];
    for (int i = threadIdx.x; i < KTAPS * C_; i += blockDim.x) {
        int k = i / C_;
        int c = i - k * C_;
        wlds[i] = weight[c * KTAPS + k];
    }
    __syncthreads();

    const int lane = threadIdx.x & 31;
    const int n    = lane & 15;   // pixel within tile (N column)
    const int half = lane >> 4;   // 0: K={0,1}, 1: K={2,3}

    const int gw = (int)((blockIdx.x * blockDim.x + threadIdx.x) >> 5);
    const int b  = gw / TILES_PER_IMG;
    const int tp = gw - b * TILES_PER_IMG;
    const int h  = tp >> 4;            // row of this tile
    const int w0 = (tp & 15) << 4;     // first pixel column of tile

    const float* fb = feature + (size_t)b * C_ * HW;

    v8f acc = {};
    for (int k = 0; k < KTAPS; ++k) {
        const int sidx = g_idx[k * HW + h * W_ + w0 + n];
        const float* fp = fb + sidx + (size_t)(2 * half) * HW;
        const float* wp = wlds + k * C_ + 2 * half;
        #pragma unroll 8
        for (int c = 0; c < C_; c += 4) {
            v2f bm;                        // B: features, channels K,K+1 @ pixel n
            bm.x = fp[0];
            bm.y = fp[HW];
            v2f am = *(const v2f*)(wp);    // A: weights (ds_load_b64, row-replicated)
            acc = __builtin_amdgcn_wmma_f32_16x16x4_f32(
                false, am, false, bm, (short)0, acc, false, false);
            fp += (size_t)4 * HW;
            wp += 4;
        }
    }

    // Every lane's acc[0] holds out[pixel = lane%16]; lanes 0-15 store the
    // 2x2 nearest-upsampled result as two 8-byte stores.
    float val = acc[0];
    if (half == 0) {
        const int h2 = h * 2;
        const int w2 = (w0 + n) * 2;
        float2 v = make_float2(val, val);
        float* ob = out + (size_t)b * (2 * H_) * (2 * W_);
        *(float2*)(ob + (size_t)h2 * (2 * W_) + w2)       = v;
        *(float2*)(ob + (size_t)(h2 + 1) * (2 * W_) + w2) = v;
    }
}

// ---------------------------------------------------------------------------
extern "C" void kernel_launch(void* const* d_in, const int* in_sizes, int n_in,
                              void* d_out, int out_size, void* d_ws, size_t ws_size,
                              hipStream_t stream) {
    const float* feature = (const float*)d_in[0]; // [4,128,128,256]
    const float* weight  = (const float*)d_in[1]; // [1,128,3,3]
    float* out = (float*)d_out;                   // [4,1,256,512]
    int* g_idx = (int*)d_ws;                      // 9*H*W ints = 1.15 MiB

    const int idx_total = KTAPS * HW;
    sphere_idx_kernel<<<(idx_total + 255) / 256, 256, 0, stream>>>(g_idx);

    // 4 imgs * 2048 tiles = 8192 waves = 262144 threads = 1024 blocks of 256
    sphere_conv_wmma<<<(B_ * TILES_PER_IMG * 32) / 256, 256, 0, stream>>>(
        feature, weight, g_idx, out);
}